// Non_local_1365799600250
// MI455X (gfx1250) — compile-verified
//
#include <hip/hip_runtime.h>

typedef float v2f __attribute__((ext_vector_type(2)));
typedef float v4f __attribute__((ext_vector_type(4)));
typedef float v8f __attribute__((ext_vector_type(8)));

#define Bn 64
#define Cn 256
#define Nn 2592
#define N4c (Nn / 4)       // 648 float4 per (b,c) row
#define TILES (Nn / 16)    // 162 16-position tiles per batch
#define WAVES_PB 6
#define GRID_X (TILES / WAVES_PB) // 27
#define EPS 1e-5f

// ---------------------------------------------------------------------------
// Kernel 0: fold BatchNorm + W conv into per-channel affine A[c], D[c]
// ---------------------------------------------------------------------------
__global__ __launch_bounds__(Cn) void nl_precomp(
    const float* __restrict__ W_w, const float* __restrict__ W_b,
    const float* __restrict__ gamma, const float* __restrict__ beta,
    const float* __restrict__ mean, const float* __restrict__ var,
    float* __restrict__ A, float* __restrict__ D)
{
    int c = threadIdx.x;
    float inv = gamma[c] / sqrtf(var[c] + EPS);
    A[c] = W_w[c] * inv;
    D[c] = W_b[c] * inv + beta[c] - mean[c] * inv;
}

// ---------------------------------------------------------------------------
// Kernel 1: channel contraction via V_WMMA_F32_16X16X4_F32.
// One wave handles a 16-position tile of one batch. A-matrix rows:
//   M=0 -> theta_w, M=1 -> phi_w, M=2 -> g_w, M=3..15 -> 0 (padding).
// A 16x4 f32 layout: lanes 0-15 hold (K=0,K=1) in (v0,v1); lanes 16-31 (K=2,K=3).
// B 4x16 f32 layout mirrors it: lane%16 = N, lane/16 selects K-half.
// C/D 16x16 f32: acc[j] on lanes 0-15 is row M=j, col N=lane.
// ---------------------------------------------------------------------------
__global__ __launch_bounds__(WAVES_PB * 32) void nl_pass1(
    const float* __restrict__ x,
    const float* __restrict__ g_w,  const float* __restrict__ g_b,
    const float* __restrict__ th_w, const float* __restrict__ th_b,
    const float* __restrict__ ph_w, const float* __restrict__ ph_b,
    float* __restrict__ theta_out, float* __restrict__ partials)
{
    const int lane = threadIdx.x & 31;
    const int wave = threadIdx.x >> 5;
    const int tile = blockIdx.x * WAVES_PB + wave;  // 0..161
    const int b    = blockIdx.y;                    // 0..63
    const int n0   = tile * 16;
    const int nl   = lane & 15;                     // col within tile / A row
    const int kb   = (lane >> 4) << 1;              // K half: 0 or 2

    // branchless A-row selection (rows >= 3 are zero padding)
    const int m = nl;
    const float* __restrict__ wrow = (m == 1) ? ph_w : ((m == 2) ? g_w : th_w);
    const float sel = (m < 3) ? 1.0f : 0.0f;

    const float* __restrict__ xb = x + (size_t)b * Cn * Nn + n0 + nl;

    v8f acc = {};
#pragma unroll 8
    for (int c = 0; c < Cn; c += 4) {
        v2f a, bm;
        a.x  = wrow[c + kb]     * sel;
        a.y  = wrow[c + kb + 1] * sel;
        bm.x = xb[(size_t)(c + kb)     * Nn];
        bm.y = xb[(size_t)(c + kb + 1) * Nn];
        acc = __builtin_amdgcn_wmma_f32_16x16x4_f32(
            /*neg_a=*/false, a, /*neg_b=*/false, bm,
            /*c_mod=*/(short)0, acc, /*reuse_a=*/false, /*reuse_b=*/false);
    }

    // rows 0/1/2 live in acc[0..2] on lanes 0-15
    const float th = acc[0] + th_b[0];
    const float ph = acc[1] + ph_b[0];
    const float gg = acc[2] + g_b[0];

    float p = (lane < 16) ? (ph * gg) : 0.0f;
#pragma unroll
    for (int off = 16; off > 0; off >>= 1)
        p += __shfl_xor(p, off, 32);

    if (lane < 16)
        theta_out[(size_t)b * Nn + n0 + nl] = th;
    if (lane == 0)
        partials[b * TILES + tile] = p;
}

// ---------------------------------------------------------------------------
// Kernel 2: deterministic reduction of per-tile partials -> s[b]
// ---------------------------------------------------------------------------
__global__ __launch_bounds__(256) void nl_reduce_s(
    const float* __restrict__ partials, float* __restrict__ s)
{
    __shared__ float sm[256];
    int b = blockIdx.x, t = threadIdx.x;
    sm[t] = (t < TILES) ? partials[b * TILES + t] : 0.0f;
    __syncthreads();
#pragma unroll
    for (int off = 128; off > 0; off >>= 1) {
        if (t < off) sm[t] += sm[t + off];
        __syncthreads();
    }
    if (t == 0) s[b] = sm[0] / (float)Nn;
}

// ---------------------------------------------------------------------------
// Kernel 3: fused epilogue z = x + A[c]*s[b]*theta[b,n] + D[c].
// x was just streamed through the 192MB L2 by pass1; NT stores for z avoid
// evicting it so the second x read mostly hits L2.
// ---------------------------------------------------------------------------
__global__ __launch_bounds__(128) void nl_pass2(
    const v4f* __restrict__ x4,
    const v4f* __restrict__ theta4,
    const float* __restrict__ s,
    const float* __restrict__ A, const float* __restrict__ D,
    v4f* __restrict__ z4)
{
    int n4 = blockIdx.x * blockDim.x + threadIdx.x;
    if (n4 >= N4c) return;
    int c = blockIdx.y, b = blockIdx.z;

    float a = A[c] * s[b];
    float d = D[c];
    v4f th = theta4[(size_t)b * N4c + n4];
    size_t idx = ((size_t)b * Cn + c) * N4c + n4;
    v4f xv = x4[idx];
    v4f zv;
    zv.x = fmaf(a, th.x, xv.x + d);
    zv.y = fmaf(a, th.y, xv.y + d);
    zv.z = fmaf(a, th.z, xv.z + d);
    zv.w = fmaf(a, th.w, xv.w + d);
    __builtin_nontemporal_store(zv, &z4[idx]);
}

// ---------------------------------------------------------------------------
extern "C" void kernel_launch(void* const* d_in, const int* in_sizes, int n_in,
                              void* d_out, int out_size, void* d_ws, size_t ws_size,
                              hipStream_t stream)
{
    const float* x     = (const float*)d_in[0];
    const float* g_w   = (const float*)d_in[1];
    const float* g_b   = (const float*)d_in[2];
    const float* th_w  = (const float*)d_in[3];
    const float* th_b  = (const float*)d_in[4];
    const float* ph_w  = (const float*)d_in[5];
    const float* ph_b  = (const float*)d_in[6];
    const float* W_w   = (const float*)d_in[7];
    const float* W_b   = (const float*)d_in[8];
    const float* bn_g  = (const float*)d_in[9];
    const float* bn_b  = (const float*)d_in[10];
    const float* bn_m  = (const float*)d_in[11];
    const float* bn_v  = (const float*)d_in[12];

    float* ws      = (float*)d_ws;
    float* theta   = ws;                         // B*N      = 165888
    float* parts   = theta + (size_t)Bn * Nn;    // B*TILES  = 10368
    float* s       = parts + (size_t)Bn * TILES; // 64
    float* Avec    = s + Bn;                     // 256
    float* Dvec    = Avec + Cn;                  // 256

    nl_precomp<<<1, Cn, 0, stream>>>(W_w, W_b, bn_g, bn_b, bn_m, bn_v, Avec, Dvec);

    nl_pass1<<<dim3(GRID_X, Bn), WAVES_PB * 32, 0, stream>>>(
        x, g_w, g_b, th_w, th_b, ph_w, ph_b, theta, parts);

    nl_reduce_s<<<Bn, 256, 0, stream>>>(parts, s);

    nl_pass2<<<dim3((N4c + 127) / 128, Cn, Bn), 128, 0, stream>>>(
        (const v4f*)x, (const v4f*)theta, s, Avec, Dvec, (v4f*)d_out);
}